// iRPE_SA_65781719105838
// MI455X (gfx1250) — compile-verified
//
#include <hip/hip_runtime.h>
#include <hip/hip_bf16.h>

// MI455X (gfx1250) wave32 / WMMA implementation.
// Bandwidth-bound (~2.0 GB traffic -> ~87us at 23.3 TB/s vs 238 GFLOP), so
// fp32 end-to-end with V_WMMA_F32_16X16X4_F32 on the matrix pipe.
// This revision: k-pair-interleaved B layouts + even LDS strides so every
// WMMA operand is one ds_load_b64 (no repack movs), batched operand loads
// feeding back-to-back WMMA bursts, float4 staging, wave-parallel softmax.

typedef __attribute__((ext_vector_type(2))) float v2f;
typedef __attribute__((ext_vector_type(8))) float v8f;

static __device__ __forceinline__ v8f wmma_f32(v2f a, v2f b, v8f c) {
  // 8 args: (neg_a, A, neg_b, B, c_mod, C, reuse_a, reuse_b)
  return __builtin_amdgcn_wmma_f32_16x16x4_f32(
      false, a, false, b, (short)0, c, false, false);
}

// Problem constants
#define BW      2048      // B*W = 32*64
#define TT      49        // tokens per window
#define EE      512
#define HH      16
#define HD      32
#define LL      169       // (2*6+1)^2
#define MROWS   100352    // BW*TT
#define BLK_ELE 1568      // TT*HD, per (bw,h) tensor slice

// LDS strides (even / multiple-of-4 => 8B/16B aligned v2f/float4 accesses)
#define AST 36            // A-side tiles (row-major, K contiguous)
#define BKP 160           // B interleaved: 160 floats per k-pair row (80 v2f)

// ---------------------------------------------------------------------------
// Shared GEMM core helpers (64x64 tile, 8 waves, 2 16x16 tiles per wave)
// As: [64][AST] row-major (K contiguous).  Bs: k-pair interleaved,
// element (k, n) at Bs[(k>>1)*BKP + n*2 + (k&1)] -> one ds_load_b64/operand.
// ---------------------------------------------------------------------------

// ---------------------------------------------------------------------------
// Stage A: qkv = x @ qkv_w + qkv_b, scattered into q/k/v [bw, h, t, hd]
// M=100352, K=512, N=1536.
// ---------------------------------------------------------------------------
__global__ __launch_bounds__(256) void qkv_proj_kernel(
    const float* __restrict__ x, const float* __restrict__ w,
    const float* __restrict__ bvec,
    float* __restrict__ qb, float* __restrict__ kb, float* __restrict__ vb) {
  __shared__ float As[64 * AST];
  __shared__ float Bs[16 * BKP];
  const int tid  = threadIdx.x;
  const int wave = tid >> 5, lane = tid & 31;
  const int r    = lane & 15, half = lane >> 4;
  const int wm   = wave & 3,  wn   = wave >> 2;
  const int m0   = blockIdx.x * 64, n0 = blockIdx.y * 64;

  v8f c0 = {}; v8f c1 = {};
  for (int k0 = 0; k0 < 512; k0 += 32) {
#pragma unroll
    for (int i = 0; i < 2; ++i) {          // A chunk: 64 rows x 32 k (float4)
      int idx4 = tid + i * 256;
      int row = idx4 >> 3, col4 = (idx4 & 7) << 2;
      const float* gp = &x[(size_t)(m0 + row) * EE + k0 + col4];
      float4 tv = *(const float4*)gp;
      if (k0 + 32 < 512) __builtin_prefetch(gp + 32, 0, 3);
      *(float4*)&As[row * AST + col4] = tv;
    }
#pragma unroll
    for (int i = 0; i < 2; ++i) {          // B chunk: 32 k x 64 n (float4)
      int idx4 = tid + i * 256;
      int row = idx4 >> 4, col4 = (idx4 & 15) << 2;
      const float* gp = &w[(size_t)(k0 + row) * 1536 + n0 + col4];
      float4 tv = *(const float4*)gp;
      if (k0 + 32 < 512) __builtin_prefetch(gp + 32 * 1536, 0, 3);
      int p = row >> 1, par = row & 1;
      float* bp = &Bs[p * BKP + col4 * 2 + par];
      bp[0] = tv.x; bp[2] = tv.y; bp[4] = tv.z; bp[6] = tv.w;
    }
    __syncthreads();

    v2f a8[8], b0r[8], b1r[8];
#pragma unroll
    for (int k8 = 0; k8 < 8; ++k8) {       // batch all operand loads
      a8[k8]  = *(const v2f*)&As[(wm * 16 + r) * AST + k8 * 4 + 2 * half];
      b0r[k8] = *(const v2f*)&Bs[(k8 * 2 + half) * BKP + (wn * 32 + r) * 2];
      b1r[k8] = *(const v2f*)&Bs[(k8 * 2 + half) * BKP + (wn * 32 + 16 + r) * 2];
    }
#pragma unroll
    for (int k8 = 0; k8 < 8; ++k8) {       // 16 back-to-back WMMAs
      c0 = wmma_f32(a8[k8], b0r[k8], c0);
      c1 = wmma_f32(a8[k8], b1r[k8], c1);
    }
    __syncthreads();
  }

#pragma unroll
  for (int v = 0; v < 8; ++v) {
    int m  = m0 + wm * 16 + v + 8 * half;
    int bw = m / TT;
    int t  = m - bw * TT;
#pragma unroll
    for (int j = 0; j < 2; ++j) {
      int   n   = n0 + wn * 32 + j * 16 + r;
      float val = (j == 0 ? c0[v] : c1[v]) + bvec[n];
      int sel = n >> 9;                // 0=q 1=k 2=v
      int hh  = (n >> 5) & 15;
      int e   = n & 31;
      float* dst = (sel == 0) ? qb : (sel == 1) ? kb : vb;
      dst[((size_t)bw * HH + hh) * BLK_ELE + (size_t)t * HD + e] = val;
    }
  }
}

// ---------------------------------------------------------------------------
// Stage B: per-(bw,h) attention. One 256-thread WG per block (32768 blocks).
// LDS: qs/ks [64x36], rpe [176x36], v^T [32x66], scores [49x177], P [64x66]
// ~101 KB dynamic LDS (gfx1250: 320 KB/WGP). Output overwrites q buffer.
// ---------------------------------------------------------------------------
#define SM_QS 0
#define SM_KS (SM_QS + 64 * AST)
#define SM_RS (SM_KS + 64 * AST)
#define SM_VT (SM_RS + 176 * AST)
#define SM_SC (SM_VT + 32 * 66)
#define SM_P  (SM_SC + 49 * 177)
#define SM_TOT (SM_P + 64 * 66)

__global__ __launch_bounds__(256) void attn_kernel(
    const float* qb, const float* __restrict__ kb,
    const float* __restrict__ vb, const float* __restrict__ rpe,
    float* ob) {
  extern __shared__ float sm[];
  float* qs = sm + SM_QS;   // q  [64][36] (rows >=49 zero)
  float* ks = sm + SM_KS;   // k  [64][36]
  float* rs = sm + SM_RS;   // rpe[176][36]
  float* vt = sm + SM_VT;   // v^T [32][66] (cols >=49 zero)
  float* sc = sm + SM_SC;   // scores [49][177]
  float* P  = sm + SM_P;    // logits -> probs [64][66]

  const int tid  = threadIdx.x;
  const int wave = tid >> 5, lane = tid & 31;
  const int r    = lane & 15, half = lane >> 4;
  const int bwh  = blockIdx.x;
  const int h    = bwh & 15;

  const float* qg = qb  + (size_t)bwh * BLK_ELE;
  const float* kg = kb  + (size_t)bwh * BLK_ELE;
  const float* vg = vb  + (size_t)bwh * BLK_ELE;
  const float* rg = rpe + (size_t)h * (LL * HD);

#pragma unroll
  for (int i = 0; i < 2; ++i) {            // q, k, v staging (float4)
    int idx4 = tid + i * 256;              // 512 float4 slots = 64 rows x 32
    int row = idx4 >> 3, col4 = (idx4 & 7) << 2;
    float4 qv = {0.f, 0.f, 0.f, 0.f}, kv = qv, vv = qv;
    if (row < TT) {
      qv = *(const float4*)&qg[idx4 * 4];
      kv = *(const float4*)&kg[idx4 * 4];
      vv = *(const float4*)&vg[idx4 * 4];
    }
    *(float4*)&qs[row * AST + col4] = qv;
    *(float4*)&ks[row * AST + col4] = kv;
    vt[(col4 + 0) * 66 + row] = vv.x;      // transpose scatter
    vt[(col4 + 1) * 66 + row] = vv.y;
    vt[(col4 + 2) * 66 + row] = vv.z;
    vt[(col4 + 3) * 66 + row] = vv.w;
  }
#pragma unroll
  for (int i = 0; i < 6; ++i) {            // rpe staging: 1408 float4 slots
    int idx4 = tid + i * 256;
    if (idx4 < 176 * 8) {
      int row = idx4 >> 3, col4 = (idx4 & 7) << 2;
      float4 tv = {0.f, 0.f, 0.f, 0.f};
      if (row < LL) tv = *(const float4*)&rg[idx4 * 4];
      *(float4*)&rs[row * AST + col4] = tv;
    }
  }
  __syncthreads();

  // Unified column jobs: job<11 -> scores col-tile (B=rpe), else qk^T col-tile
  // (B=k). 4 row-tile accumulators share each B operand load.
  const float scale = 0.17677669529663688f;  // hd^-0.5
  for (int job = wave; job < 15; job += 8) {
    const bool qk = (job >= 11);
    const int  nt = qk ? (job - 11) : job;
    const float* bbase = (qk ? ks : rs) + (nt * 16 + r) * AST;
    v8f ac0 = {}, ac1 = {}, ac2 = {}, ac3 = {};
#pragma unroll
    for (int k8 = 0; k8 < 8; ++k8) {
      const int ko = k8 * 4 + 2 * half;
      v2f b  = *(const v2f*)&bbase[ko];
      v2f a0 = *(const v2f*)&qs[(0  + r) * AST + ko];
      v2f a1 = *(const v2f*)&qs[(16 + r) * AST + ko];
      v2f a2 = *(const v2f*)&qs[(32 + r) * AST + ko];
      v2f a3 = *(const v2f*)&qs[(48 + r) * AST + ko];
      ac0 = wmma_f32(a0, b, ac0);
      ac1 = wmma_f32(a1, b, ac1);
      ac2 = wmma_f32(a2, b, ac2);
      ac3 = wmma_f32(a3, b, ac3);
    }
    const int col = nt * 16 + r;
    if (qk) {
#pragma unroll
      for (int v = 0; v < 8; ++v) {
        int rr = v + 8 * half;
        P[(rr +  0) * 66 + col] = ac0[v] * scale;
        P[(rr + 16) * 66 + col] = ac1[v] * scale;
        P[(rr + 32) * 66 + col] = ac2[v] * scale;
        P[(rr + 48) * 66 + col] = ac3[v] * scale;
      }
    } else if (col < LL) {
#pragma unroll
      for (int v = 0; v < 8; ++v) {
        int rr = v + 8 * half;
        sc[(rr +  0) * 177 + col] = ac0[v];
        sc[(rr + 16) * 177 + col] = ac1[v];
        sc[(rr + 32) * 177 + col] = ac2[v];
        if (rr + 48 < TT) sc[(rr + 48) * 177 + col] = ac3[v];
      }
    }
  }
  __syncthreads();

  // Wave-parallel softmax with RPE bias gather: one wave per row,
  // lanes cover s={lane, lane+32}, shfl_xor reductions.
  for (int row = wave; row < 64; row += 8) {
    const int s0 = lane, s1 = lane + 32;
    if (row < TT) {
      const int ty = row / 7, tx = row % 7;
      int sy = s0 / 7, sx = s0 % 7;
      int dy = ty - sy; dy = dy < -6 ? -6 : (dy > 6 ? 6 : dy);
      int dx = tx - sx; dx = dx < -6 ? -6 : (dx > 6 ? 6 : dx);
      float l0 = P[row * 66 + s0] + sc[row * 177 + (dy + 6) * 13 + (dx + 6)];
      float l1 = -3.0e38f;
      if (s1 < TT) {
        sy = s1 / 7; sx = s1 % 7;
        dy = ty - sy; dy = dy < -6 ? -6 : (dy > 6 ? 6 : dy);
        dx = tx - sx; dx = dx < -6 ? -6 : (dx > 6 ? 6 : dx);
        l1 = P[row * 66 + s1] + sc[row * 177 + (dy + 6) * 13 + (dx + 6)];
      }
      float m = fmaxf(l0, l1);
#pragma unroll
      for (int off = 16; off > 0; off >>= 1)
        m = fmaxf(m, __shfl_xor(m, off, 32));
      float e0 = __expf(l0 - m);
      float e1 = (s1 < TT) ? __expf(l1 - m) : 0.f;
      float sum = e0 + e1;
#pragma unroll
      for (int off = 16; off > 0; off >>= 1)
        sum += __shfl_xor(sum, off, 32);
      float inv = 1.0f / sum;
      P[row * 66 + s0] = e0 * inv;
      P[row * 66 + s1] = e1 * inv;   // zero for s1 >= 49
    } else {
      P[row * 66 + s0] = 0.f;
      P[row * 66 + s1] = 0.f;
    }
  }
  __syncthreads();

  // O = P @ v (K=64, zero-padded); one 16x16 tile per wave; overwrite q buf.
  {
    const int mt = wave & 3, nt = wave >> 2;
    v8f acc = {};
    v2f ar[16], br[16];
#pragma unroll
    for (int k8 = 0; k8 < 16; ++k8) {
      const int ko = k8 * 4 + 2 * half;
      ar[k8] = *(const v2f*)&P[(mt * 16 + r) * 66 + ko];
      br[k8] = *(const v2f*)&vt[(nt * 16 + r) * 66 + ko];  // B[k][c] = v^T[c][k]
    }
#pragma unroll
    for (int k8 = 0; k8 < 16; ++k8) acc = wmma_f32(ar[k8], br[k8], acc);
#pragma unroll
    for (int v = 0; v < 8; ++v) {
      int row = mt * 16 + v + 8 * half;
      int col = nt * 16 + r;
      if (row < TT)
        ob[(size_t)bwh * BLK_ELE + (size_t)row * HD + col] = acc[v];
    }
  }
}

// ---------------------------------------------------------------------------
// Stage C: out = attn_out @ proj_w + proj_b. A gathered from [bw,h,t,hd].
// M=100352, K=512, N=512.
// ---------------------------------------------------------------------------
__global__ __launch_bounds__(256) void out_proj_kernel(
    const float* __restrict__ abuf, const float* __restrict__ w,
    const float* __restrict__ bvec, float* __restrict__ out) {
  __shared__ float As[64 * AST];
  __shared__ float Bs[16 * BKP];
  const int tid  = threadIdx.x;
  const int wave = tid >> 5, lane = tid & 31;
  const int r    = lane & 15, half = lane >> 4;
  const int wm   = wave & 3,  wn   = wave >> 2;
  const int m0   = blockIdx.x * 64, n0 = blockIdx.y * 64;

  v8f c0 = {}; v8f c1 = {};
  for (int k0 = 0; k0 < 512; k0 += 32) {
    const int hh = k0 >> 5;                // one head per 32-wide k-chunk
#pragma unroll
    for (int i = 0; i < 2; ++i) {
      int idx4 = tid + i * 256;
      int row = idx4 >> 3, col4 = (idx4 & 7) << 2;
      int m = m0 + row;
      int bw = m / TT;
      int t  = m - bw * TT;
      float4 tv = *(const float4*)
          &abuf[((size_t)bw * HH + hh) * BLK_ELE + (size_t)t * HD + col4];
      *(float4*)&As[row * AST + col4] = tv;
    }
#pragma unroll
    for (int i = 0; i < 2; ++i) {
      int idx4 = tid + i * 256;
      int row = idx4 >> 4, col4 = (idx4 & 15) << 2;
      const float* gp = &w[(size_t)(k0 + row) * EE + n0 + col4];
      float4 tv = *(const float4*)gp;
      if (k0 + 32 < 512) __builtin_prefetch(gp + 32 * EE, 0, 3);
      int p = row >> 1, par = row & 1;
      float* bp = &Bs[p * BKP + col4 * 2 + par];
      bp[0] = tv.x; bp[2] = tv.y; bp[4] = tv.z; bp[6] = tv.w;
    }
    __syncthreads();

    v2f a8[8], b0r[8], b1r[8];
#pragma unroll
    for (int k8 = 0; k8 < 8; ++k8) {
      a8[k8]  = *(const v2f*)&As[(wm * 16 + r) * AST + k8 * 4 + 2 * half];
      b0r[k8] = *(const v2f*)&Bs[(k8 * 2 + half) * BKP + (wn * 32 + r) * 2];
      b1r[k8] = *(const v2f*)&Bs[(k8 * 2 + half) * BKP + (wn * 32 + 16 + r) * 2];
    }
#pragma unroll
    for (int k8 = 0; k8 < 8; ++k8) {
      c0 = wmma_f32(a8[k8], b0r[k8], c0);
      c1 = wmma_f32(a8[k8], b1r[k8], c1);
    }
    __syncthreads();
  }

#pragma unroll
  for (int v = 0; v < 8; ++v) {
    int m = m0 + wm * 16 + v + 8 * half;
#pragma unroll
    for (int j = 0; j < 2; ++j) {
      int   n   = n0 + wn * 32 + j * 16 + r;
      float val = (j == 0 ? c0[v] : c1[v]) + bvec[n];
      out[(size_t)m * EE + n] = val;
    }
  }
}

// ---------------------------------------------------------------------------
extern "C" void kernel_launch(void* const* d_in, const int* in_sizes, int n_in,
                              void* d_out, int out_size, void* d_ws,
                              size_t ws_size, hipStream_t stream) {
  (void)in_sizes; (void)n_in; (void)out_size; (void)ws_size;
  const float* x      = (const float*)d_in[0];
  const float* qkv_w  = (const float*)d_in[1];
  const float* qkv_b  = (const float*)d_in[2];
  const float* proj_w = (const float*)d_in[3];
  const float* proj_b = (const float*)d_in[4];
  const float* rpe    = (const float*)d_in[5];

  // Workspace: q/k/v in [bw, h, t, hd] layout, 205.5 MB each (616.6 MB total).
  // Attention output overwrites the q buffer.
  float* ws = (float*)d_ws;
  const size_t perBuf = (size_t)BW * HH * BLK_ELE;  // 51,380,224 floats
  float* qb = ws;
  float* kb = ws + perBuf;
  float* vb = ws + 2 * perBuf;

  dim3 blk(256);
  qkv_proj_kernel<<<dim3(MROWS / 64, 1536 / 64), blk, 0, stream>>>(
      x, qkv_w, qkv_b, qb, kb, vb);
  attn_kernel<<<dim3(BW * HH), blk, SM_TOT * sizeof(float), stream>>>(
      qb, kb, vb, rpe, qb);
  out_proj_kernel<<<dim3(MROWS / 64, EE / 64), blk, 0, stream>>>(
      qb, proj_w, proj_b, (float*)d_out);
}